// getTargets_24421184045192
// MI455X (gfx1250) — compile-verified
//
#include <hip/hip_runtime.h>
#include <cstdint>
#include <cstddef>

// ---- problem constants (from reference setup_inputs) ----
#define MODEL_W 1280.0f
#define MODEL_H 768.0f
#define GH 384          // MODEL_H // 2
#define GW 640          // MODEL_W // 2
#define BS 4
#define NB 64

typedef __attribute__((ext_vector_type(2)))  float        v2f;
typedef __attribute__((ext_vector_type(8)))  float        v8f;
typedef __attribute__((ext_vector_type(4)))  unsigned int u32x4;
typedef __attribute__((ext_vector_type(8)))  int          i32x8;
typedef __attribute__((ext_vector_type(4)))  int          i32x4;

// ---------------- shared helpers ----------------
struct GtBox {
  float gminx, gminy, gmaxx, gmaxy, area;
  float w0, w1, h0, h1;   // region bounds (exact integers stored as float)
  bool  valid;
};

__device__ __forceinline__ GtBox make_gt(const float* bx) {
  GtBox g;
  float x1 = bx[0], y1 = bx[1], x2 = bx[2], y2 = bx[3];
  float bw = x2 - x1, bh = y2 - y1;
  g.valid = (bw * bh) != 0.0f;
  g.area  = bw * bh;
  float cx = x1 + 0.5f * bw, cy = y1 + 0.5f * bh;
  g.gminx = cx - 0.5f * bw; g.gmaxx = cx + 0.5f * bw;
  g.gminy = cy - 0.5f * bh; g.gmaxy = cy + 0.5f * bh;
  g.w0 = floorf(fmaxf(x1 * ((float)GW / MODEL_W) - 0.5f, 0.0f));
  g.h0 = floorf(fmaxf(y1 * ((float)GH / MODEL_H) - 0.5f, 0.0f));
  g.w1 = ceilf (fminf(x2 * ((float)GW / MODEL_W) - 0.5f, (float)GW - 1.0f));
  g.h1 = ceilf (fminf(y2 * ((float)GH / MODEL_H) - 0.5f, (float)GH - 1.0f));
  return g;
}

__device__ __forceinline__ float iou_px(float p0, float p1, float p2, float p3,
                                        int h, int w, const GtBox& g) {
  float rx = ((float)w + 0.5f) * (MODEL_W / (float)GW);
  float ry = ((float)h + 0.5f) * (MODEL_H / (float)GH);
  float x1 = p0 * 80.0f + rx, y1 = p1 * 80.0f + ry;
  float x2 = p2 * 80.0f + rx, y2 = p3 * 80.0f + ry;
  float pw = x2 - x1, ph = y2 - y1;
  float cx = x1 + 0.5f * pw, cy = y1 + 0.5f * ph;
  float pminx = cx - 0.5f * pw, pmaxx = cx + 0.5f * pw;
  float pminy = cy - 0.5f * ph, pmaxy = cy + 0.5f * ph;
  float iw = fmaxf(fminf(pmaxx, g.gmaxx) - fmaxf(pminx, g.gminx), 0.0f);
  float ih = fmaxf(fminf(pmaxy, g.gmaxy) - fmaxf(pminy, g.gminy), 0.0f);
  float inter = iw * ih;
  return inter / fmaxf(pw * ph + g.area - inter, 1e-6f);
}

// 32-lane sum. CDNA5 path: row-sum-by-ones via v_wmma_f32_16x16x4_f32.
// A[m][k]: lane L holds A[L%16][k0],A[L%16][k0+1], k0=(L<16)?0:2 -> a={x,0}
// gives rowsum[m] = x[m] + x[m+16]; D rows 0..7 land in lanes 0..15 (c[0..7]),
// rows 8..15 in lanes 16..31.  One xor-16 shuffle finishes the sum.
__device__ __forceinline__ float wave_reduce_sum(float x) {
#if defined(__HIP_DEVICE_COMPILE__) && __has_builtin(__builtin_amdgcn_wmma_f32_16x16x4_f32)
  v2f a; a[0] = x;    a[1] = 0.0f;
  v2f b; b[0] = 1.0f; b[1] = 1.0f;
  v8f c = {};
  c = __builtin_amdgcn_wmma_f32_16x16x4_f32(false, a, false, b, (short)0, c, false, false);
  float s = c[0] + c[1] + c[2] + c[3] + c[4] + c[5] + c[6] + c[7];
  s += __shfl_xor(s, 16, 32);
  return s;
#else
  for (int o = 16; o > 0; o >>= 1) x += __shfl_xor(x, o, 32);
  return x;
#endif
}

// ---------------- kernel 1: per-box region-masked IoU sums -> dk ----------------
__global__ __launch_bounds__(256) void kBoxSums(const float* __restrict__ pred,
                                                const float* __restrict__ boxes,
                                                float* __restrict__ sums) {
  const int bn = blockIdx.x, b = bn / NB, n = bn % NB;
  GtBox g = make_gt(boxes + ((size_t)b * NB + n) * 6);
  const int w0 = (int)g.w0, w1 = (int)g.w1, h0 = (int)g.h0, h1 = (int)g.h1;
  float acc = 0.0f;
  if (g.valid && w1 >= w0 && h1 >= h0) {
    const int rw = w1 - w0 + 1, cnt = rw * (h1 - h0 + 1);
    const size_t plane = (size_t)GH * GW;
    const float* base = pred + (size_t)b * 4 * plane;
    for (int i = threadIdx.x; i < cnt; i += blockDim.x) {
      int h = h0 + i / rw, w = w0 + i % rw;
      size_t off = (size_t)h * GW + w;
      __builtin_prefetch(base + off + GW, 0, 1);   // -> global_prefetch_b8
      acc += iou_px(base[off], base[plane + off], base[2 * plane + off],
                    base[3 * plane + off], h, w, g);
    }
  }
  float wsum = wave_reduce_sum(acc);               // uniform path: EXEC all-ones
  __shared__ float bsum;
  if (threadIdx.x == 0) bsum = 0.0f;
  __syncthreads();
  if ((threadIdx.x & 31) == 0) atomicAdd(&bsum, wsum);   // ds_add_f32
  __syncthreads();
  if (threadIdx.x == 0) sums[bn] = bsum;
}

// ---------------- kernel 2: per-box 4-pass radix select of sorted_desc[k] ----------------
// iou_f >= 0 so float bit patterns order like uint32.  Values outside the region
// are implicit zeros; if the rank falls past all histogrammed elements every pass
// picks bucket 0 and the selected value is 0 — exactly sorted_desc[k].
__global__ __launch_bounds__(256) void kRadixSelect(const float* __restrict__ pred,
                                                    const float* __restrict__ boxes,
                                                    const float* __restrict__ sums,
                                                    float* __restrict__ thrs) {
  __shared__ unsigned hist[256];
  __shared__ unsigned s_pfx, s_rank;
  const int bn = blockIdx.x, b = bn / NB, n = bn % NB;
  GtBox g = make_gt(boxes + ((size_t)b * NB + n) * 6);
  const int w0 = (int)g.w0, w1 = (int)g.w1, h0 = (int)g.h0, h1 = (int)g.h1;
  const bool nonempty = g.valid && w1 >= w0 && h1 >= h0;
  const int rw  = w1 - w0 + 1;
  const int cnt = nonempty ? rw * (h1 - h0 + 1) : 0;
  const size_t plane = (size_t)GH * GW;
  const float* base = pred + (size_t)b * 4 * plane;
  if (threadIdx.x == 0) {
    int dk = (int)ceilf(fmaxf(sums[bn], 1.0f));
    s_rank = (unsigned)min(dk, GH * GW - 1);
    s_pfx  = 0u;
  }
  for (int pass = 0; pass < 4; ++pass) {
    for (int i = threadIdx.x; i < 256; i += blockDim.x) hist[i] = 0u;
    __syncthreads();
    const unsigned pfx = s_pfx;
    const int shift = 24 - 8 * pass;
    for (int i = threadIdx.x; i < cnt; i += blockDim.x) {
      int h = h0 + i / rw, w = w0 + i % rw;
      size_t off = (size_t)h * GW + w;
      float v = iou_px(base[off], base[plane + off], base[2 * plane + off],
                       base[3 * plane + off], h, w, g);
      unsigned key = __float_as_uint(v);
      if (pass == 0 || (key >> (shift + 8)) == pfx)
        atomicAdd(&hist[(key >> shift) & 255u], 1u);
    }
    __syncthreads();
    if (threadIdx.x == 0) {
      unsigned cum = 0u, r = s_rank; int sel = 0;
      for (int i = 255; i >= 1; --i) {
        unsigned c = hist[i];
        if (cum + c > r) { sel = i; break; }
        cum += c;
      }
      s_pfx  = (pfx << 8) | (unsigned)sel;
      s_rank = r - cum;
    }
    __syncthreads();
  }
  if (threadIdx.x == 0) thrs[bn] = __uint_as_float(s_pfx);
}

// ---------------- kernel 3: per-pixel assignment ----------------
__global__ __launch_bounds__(256) void kAssign(const float* __restrict__ pred,
                                               const float* __restrict__ boxes,
                                               const float* __restrict__ sums,
                                               const float* __restrict__ thrs,
                                               const int* __restrict__ dmode,
                                               float* __restrict__ out) {
  __shared__ float sb[NB * 6];                                 // raw boxes (TDM dest)
  __shared__ float sgminx[NB], sgminy[NB], sgmaxx[NB], sgmaxy[NB], sarea[NB];
  __shared__ float sw0[NB], sw1[NB], sh0[NB], sh1[NB];
  __shared__ float sthr[NB], sscore[NB], slam[NB];
  __shared__ float scx[NB], scy[NB], sbw[NB], sbh[NB];
  __shared__ int   scls[NB], svalid[NB];

  const int bpb = (GH * GW) / 256;          // 960 blocks per batch
  const int b   = blockIdx.x / bpb;
  const int pix = (blockIdx.x % bpb) * 256 + threadIdx.x;
  const float* gb = boxes + (size_t)b * NB * 6;

#if defined(__HIP_DEVICE_COMPILE__) && __has_builtin(__builtin_amdgcn_tensor_load_to_lds) && __has_builtin(__builtin_amdgcn_s_wait_tensorcnt)
  // TDM: DMA the 64x6 f32 box table (1536 B) into LDS.  Issue from wave 0 only
  // (uniform branch -> EXEC all-ones in that wave), wait TENSORcnt, then barrier.
  if (threadIdx.x < 32) {
    unsigned           lds = (unsigned)(uintptr_t)(&sb[0]);
    unsigned long long ga  = (unsigned long long)(uintptr_t)gb;
    u32x4 g0;
    g0[0] = 1u;                                             // count=1 valid descriptor
    g0[1] = lds;                                            // lds_addr
    g0[2] = (unsigned)(ga & 0xffffffffu);                   // global_addr[31:0]
    g0[3] = (unsigned)((ga >> 32) & 0x01ffffffu) | (2u << 30); // addr[56:32] | type=2
    i32x8 g1;
    g1[0] = (int)(2u << 16);                                // data_size = 4 bytes
    g1[1] = (int)((unsigned)(NB * 6) << 16);                // tensor_dim0 = 384 (low16)
    g1[2] = (int)(1u << 16);                                // tensor_dim1 = 1  (low16)
    g1[3] = (int)((unsigned)(NB * 6) << 16);                // tile_dim0   = 384
    g1[4] = 0;                                              // tile_dim1/2 unused
    g1[5] = NB * 6;                                         // tensor_dim0_stride lo32
    g1[6] = 0; g1[7] = 0;
    i32x4 g2 = {0, 0, 0, 0};                                // dims 2/3 unused (1-D tile)
    i32x4 g3 = {0, 0, 0, 0};
    i32x8 g4 = {0, 0, 0, 0, 0, 0, 0, 0};                    // extension group (unused)
    __builtin_amdgcn_tensor_load_to_lds(g0, g1, g2, g3, g4, 0);
    __builtin_amdgcn_s_wait_tensorcnt(0);
  }
#else
  for (int i = threadIdx.x; i < NB * 6; i += blockDim.x) sb[i] = gb[i];
#endif
  __syncthreads();

  if (threadIdx.x < NB) {
    int n = threadIdx.x;
    GtBox g = make_gt(&sb[n * 6]);
    sgminx[n] = g.gminx; sgminy[n] = g.gminy; sgmaxx[n] = g.gmaxx; sgmaxy[n] = g.gmaxy;
    sarea[n] = g.area;
    sw0[n] = g.w0; sw1[n] = g.w1; sh0[n] = g.h0; sh1[n] = g.h1;
    svalid[n] = g.valid ? 1 : 0;
    float x1 = sb[n * 6 + 0], y1 = sb[n * 6 + 1], x2 = sb[n * 6 + 2], y2 = sb[n * 6 + 3];
    float bw = x2 - x1, bh = y2 - y1;
    scx[n] = x1 + 0.5f * bw; scy[n] = y1 + 0.5f * bh; sbw[n] = bw; sbh[n] = bh;
    int bn = b * NB + n;
    sthr[n] = thrs[bn];
    int dk  = (int)ceilf(fmaxf(sums[bn], 1.0f));
    slam[n] = rsqrtf((float)dk);                       // dk ** -0.5
    scls[n] = (int)sb[n * 6 + 4] + 1;
    sscore[n] = (dmode[0] != 0) ? ((sb[n * 6 + 5] >= 0.625f) ? 1.0f : 0.0f) : 1.0f;
  }
  __syncthreads();

  const int h = pix / GW, w = pix % GW;
  const size_t plane = (size_t)GH * GW;
  const float* base = pred + (size_t)b * 4 * plane + (size_t)pix;
  float p0 = base[0], p1 = base[plane], p2 = base[2 * plane], p3 = base[3 * plane];
  // decode this pixel's predicted box once
  float rx = ((float)w + 0.5f) * (MODEL_W / (float)GW);
  float ry = ((float)h + 0.5f) * (MODEL_H / (float)GH);
  float x1 = p0 * 80.0f + rx, y1 = p1 * 80.0f + ry;
  float x2 = p2 * 80.0f + rx, y2 = p3 * 80.0f + ry;
  float pw = x2 - x1, ph = y2 - y1;
  float cx = x1 + 0.5f * pw, cy = y1 + 0.5f * ph;
  float pminx = cx - 0.5f * pw, pmaxx = cx + 0.5f * pw;
  float pminy = cy - 0.5f * ph, pmaxy = cy + 0.5f * ph;
  float parea = pw * ph;
  const float fw = (float)w, fh = (float)h;

  float best = 0.0f; int winner = 0, eqcnt = 0; bool inreg = false;
  for (int n = 0; n < NB; ++n) {
    bool region = svalid[n] && fw >= sw0[n] && fw <= sw1[n] && fh >= sh0[n] && fh <= sh1[n];
    inreg = inreg || region;
    float v = 0.0f;
    if (region) {
      float iw = fmaxf(fminf(pmaxx, sgmaxx[n]) - fmaxf(pminx, sgminx[n]), 0.0f);
      float ih = fmaxf(fminf(pmaxy, sgmaxy[n]) - fmaxf(pminy, sgminy[n]), 0.0f);
      float inter = iw * ih;
      v = inter / fmaxf(parea + sarea[n] - inter, 1e-6f);
    }
    float kept = (v > sthr[n]) ? v : 0.0f;
    if (kept > best)                        { best = kept; winner = n; eqcnt = 1; }
    else if (kept > 0.0f && kept == best)   { ++eqcnt; }
  }
  const bool assigned = best > 0.0f;
  const bool tie      = eqcnt > 1;

  float c0, c1, c2;
  if (assigned) {
    if (tie) { c0 = c1 = c2 = 0.0f; }
    else {
      int ci = scls[winner];
      c0 = (ci == 0) ? 1.0f : 0.0f;
      c1 = (ci == 1) ? 1.0f : 0.0f;
      c2 = (ci == 2) ? 1.0f : 0.0f;
    }
  } else if (inreg) { c0 = c1 = c2 = 0.0f; }
  else              { c0 = 1.0f; c1 = 0.0f; c2 = 0.0f; }

  size_t co = ((size_t)b * plane + (size_t)pix) * 3u;
  out[co + 0] = c0; out[co + 1] = c1; out[co + 2] = c2;

  size_t po = (size_t)BS * plane * 3u + ((size_t)b * plane + (size_t)pix) * 6u;
  if (assigned) {
    out[po + 0] = scx[winner]; out[po + 1] = scy[winner];
    out[po + 2] = sbw[winner]; out[po + 3] = sbh[winner];
    out[po + 4] = sscore[winner]; out[po + 5] = slam[winner];
  } else {
    out[po + 0] = 1.0f; out[po + 1] = 1.0f; out[po + 2] = 1.0f;
    out[po + 3] = 1.0f; out[po + 4] = 1.0f; out[po + 5] = 1.0f;
  }
}

// ---------------- host entry ----------------
extern "C" void kernel_launch(void* const* d_in, const int* in_sizes, int n_in,
                              void* d_out, int out_size, void* d_ws, size_t ws_size,
                              hipStream_t stream) {
  (void)in_sizes; (void)n_in; (void)out_size; (void)ws_size;
  // inputs: 0=feat (unused by reference), 1=pred, 2=bboxes, 3=difficult_mode
  const float* pred  = (const float*)d_in[1];
  const float* boxes = (const float*)d_in[2];
  const int*   dmode = (const int*)d_in[3];
  float* out  = (float*)d_out;
  float* sums = (float*)d_ws;          // BS*NB floats
  float* thrs = sums + BS * NB;        // BS*NB floats (ws use: 2 KB total)

  kBoxSums   <<<BS * NB,            256, 0, stream>>>(pred, boxes, sums);
  kRadixSelect<<<BS * NB,           256, 0, stream>>>(pred, boxes, sums, thrs);
  kAssign    <<<(BS * GH * GW)/256, 256, 0, stream>>>(pred, boxes, sums, thrs, dmode, out);
}